// LigandGNN_54193897341258
// MI455X (gfx1250) — compile-verified
//
#include <hip/hip_runtime.h>
#include <hip/hip_bf16.h>

typedef float v2f __attribute__((ext_vector_type(2)));
typedef float v8f __attribute__((ext_vector_type(8)));

#define HID 128

// ---------------------------------------------------------------- utilities

__global__ void zero_f32(float* __restrict__ p, long n) {
  long i = (long)blockIdx.x * blockDim.x + threadIdx.x;
  long stride = (long)gridDim.x * blockDim.x;
  for (; i < n; i += stride) p[i] = 0.0f;
}

__global__ void init_deg(float* __restrict__ deg, int n) {
  int i = blockIdx.x * blockDim.x + threadIdx.x;
  if (i < n) deg[i] = 1.0f;  // self-loop contributes 1
}

__global__ void deg_accum(const int* __restrict__ dst, int E,
                          float* __restrict__ deg) {
  int e = blockIdx.x * blockDim.x + threadIdx.x;
  if (e < E) atomicAdd(&deg[dst[e]], 1.0f);
}

__global__ void deg_to_dinv(float* __restrict__ deg, int n) {
  int i = blockIdx.x * blockDim.x + threadIdx.x;
  if (i < n) deg[i] = rsqrtf(deg[i]);  // deg >= 1 always (self-loops)
}

// ------------------------------------------------- layer 1 linear (K = 4)

__global__ void lin_small(const float* __restrict__ x,
                          const float* __restrict__ W, float* __restrict__ h,
                          int n, int inC) {
  long tid = (long)blockIdx.x * blockDim.x + threadIdx.x;
  if (tid >= (long)n * HID) return;
  int i = (int)(tid / HID);
  int c = (int)(tid % HID);
  const float* xr = x + (long)i * inC;
  float acc = 0.0f;
  for (int k = 0; k < inC; ++k) acc += xr[k] * W[(long)k * HID + c];
  h[tid] = acc;
}

// ------------------------------------------- edge scatter (one wave/edge)

__global__ void edge_scatter(const int* __restrict__ src,
                             const int* __restrict__ dst,
                             const float* __restrict__ dinv,
                             const float* __restrict__ h,
                             float* __restrict__ out, int E) {
  long tid = (long)blockIdx.x * blockDim.x + threadIdx.x;
  long e = tid >> 5;           // one wave32 per edge
  int q = (int)(tid & 31);     // lane handles 4 of 128 channels
  if (e >= E) return;
  int s = src[e];
  int d = dst[e];
  float norm = dinv[s] * dinv[d];
  const float4* hv = (const float4*)(h + (long)s * HID);
  float4 v = hv[q];
  float* o = out + (long)d * HID + q * 4;
  atomicAdd(o + 0, v.x * norm);
  atomicAdd(o + 1, v.y * norm);
  atomicAdd(o + 2, v.z * norm);
  atomicAdd(o + 3, v.w * norm);
}

// self-loop term + bias + relu: out = relu(acc + selfh*dinv^2 + bias)
__global__ void finalize(const float* __restrict__ acc,
                         const float* __restrict__ selfh,
                         const float* __restrict__ dinv,
                         const float* __restrict__ bias,
                         float* __restrict__ out, int n) {
  long tid = (long)blockIdx.x * blockDim.x + threadIdx.x;
  if (tid >= (long)n * HID) return;
  int i = (int)(tid / HID);
  int c = (int)(tid % HID);
  float di = dinv[i];
  float v = acc[tid] + selfh[tid] * di * di + bias[c];
  out[tid] = fmaxf(v, 0.0f);
}

// ------------------------------- layer 2 GEMM: D[n,128] = A[n,128] @ W[128,128]
// fp32 WMMA: V_WMMA_F32_16X16X4_F32 (exact fp32 math, matches reference).
// Block = 256 threads = 8 waves. Block covers 16 rows; wave w covers cols
// [16w, 16w+16). A tile (16x128, 8KB) staged in LDS, shared by all waves.
// A-frag lane layout (ISA 7.12.2): M = lane%16, K-pair = k0 + 2*(lane/16).
// B-frag: N = lane%16, K-pair = k0 + 2*(lane/16).
// D: col = lane%16, rows = 8*(lane/16) + r, r = 0..7 in accum VGPRs.

__global__ void __launch_bounds__(256) gemm_wmma_f32(
    const float* __restrict__ A, const float* __restrict__ W,
    float* __restrict__ D, int n) {
  __shared__ float As[16 * HID];

  int t = threadIdx.x;
  long rowBase = (long)blockIdx.x * 16;

  // warm the cache with W (64KB, reused by every block)
  for (int line = t; line < (HID * HID) / 32; line += 256)
    __builtin_prefetch(&W[line * 32], 0, 0);

  // stage A tile: thread t loads row t/16, 8 floats at col (t%16)*8
  {
    int r = t >> 4;
    int chunk = t & 15;
    long grow = rowBase + r;
    float4 v0 = {0.f, 0.f, 0.f, 0.f}, v1 = {0.f, 0.f, 0.f, 0.f};
    if (grow < n) {
      const float* sp = A + grow * HID + chunk * 8;
      v0 = *(const float4*)(sp);
      v1 = *(const float4*)(sp + 4);
    }
    *(float4*)&As[r * HID + chunk * 8] = v0;
    *(float4*)&As[r * HID + chunk * 8 + 4] = v1;
  }
  __syncthreads();

  int wave = t >> 5;
  int lane = t & 31;
  int colBase = wave * 16;
  int half = lane >> 4;   // 0: K even pair low, 1: high
  int nn = lane & 15;     // N (B/D col) or M (A row)

  v8f c = {};
#pragma unroll
  for (int k0 = 0; k0 < HID; k0 += 4) {
    int k = k0 + 2 * half;
    v2f a;
    a.x = As[nn * HID + k];          // contiguous pair -> ds_load_b64
    a.y = As[nn * HID + k + 1];
    v2f b;
    b.x = W[(long)k * HID + colBase + nn];
    b.y = W[(long)(k + 1) * HID + colBase + nn];
    c = __builtin_amdgcn_wmma_f32_16x16x4_f32(
        /*neg_a=*/false, a, /*neg_b=*/false, b,
        /*c_mod=*/(short)0, c, /*reuse_a=*/false, /*reuse_b=*/false);
  }

  int col = colBase + nn;
  int mBase = half * 8;
#pragma unroll
  for (int r = 0; r < 8; ++r) {
    long grow = rowBase + mBase + r;
    if (grow < n) D[grow * HID + col] = c[r];
  }
}

// ------------------------------------------------------------------ pooling

__global__ void pool_cnt(const int* __restrict__ batch, float* __restrict__ cnt,
                         int n) {
  int i = blockIdx.x * blockDim.x + threadIdx.x;
  if (i < n) atomicAdd(&cnt[batch[i]], 1.0f);
}

__global__ void pool_accum(const float* __restrict__ h,
                           const int* __restrict__ batch,
                           float* __restrict__ out, int n) {
  long tid = (long)blockIdx.x * blockDim.x + threadIdx.x;
  if (tid >= (long)n * HID) return;
  int i = (int)(tid / HID);
  int c = (int)(tid % HID);
  atomicAdd(&out[(long)batch[i] * HID + c], h[tid]);
}

__global__ void pool_scale(float* __restrict__ out,
                           const float* __restrict__ cnt, int total) {
  int tid = blockIdx.x * blockDim.x + threadIdx.x;
  if (tid < total) out[tid] /= fmaxf(cnt[tid / HID], 1.0f);
}

// ------------------------------------------------------------------- launch

extern "C" void kernel_launch(void* const* d_in, const int* in_sizes, int n_in,
                              void* d_out, int out_size, void* d_ws,
                              size_t ws_size, hipStream_t stream) {
  const float* x = (const float*)d_in[0];
  const int* ei = (const int*)d_in[1];     // [2, E] (JAX int64 -> int32)
  const int* batch = (const int*)d_in[2];  // [N]
  const float* W1 = (const float*)d_in[3];
  const float* b1 = (const float*)d_in[4];
  const float* W2 = (const float*)d_in[5];
  const float* b2 = (const float*)d_in[6];

  const int N = in_sizes[2];
  const int E = in_sizes[1] / 2;
  const int inC = in_sizes[0] / N;
  const int* src = ei;
  const int* dst = ei + E;
  float* out = (float*)d_out;

  float* ws = (float*)d_ws;
  float* dinv = ws;                          // [N]
  float* bufA = dinv + N;                    // [N*HID]
  float* bufB = bufA + (long)N * HID;        // [N*HID]
  float* cnt = bufB + (long)N * HID;         // [B]

  const int T = 256;
  long nh = (long)N * HID;
  int gN = (N + T - 1) / T;
  int gE = (E + T - 1) / T;
  int gNH = (int)((nh + T - 1) / T);
  int gEdge = (int)(((long)E * 32 + T - 1) / T);
  int gGemm = (N + 15) / 16;

  // symmetric degree normalization
  init_deg<<<gN, T, 0, stream>>>(dinv, N);
  deg_accum<<<gE, T, 0, stream>>>(dst, E, dinv);
  deg_to_dinv<<<gN, T, 0, stream>>>(dinv, N);

  // layer 1: linear -> propagate -> bias+relu   (bufB = x2)
  lin_small<<<gNH, T, 0, stream>>>(x, W1, bufA, N, inC);
  zero_f32<<<gNH, T, 0, stream>>>(bufB, nh);
  edge_scatter<<<gEdge, T, 0, stream>>>(src, dst, dinv, bufA, bufB, E);
  finalize<<<gNH, T, 0, stream>>>(bufB, bufA, dinv, b1, bufB, N);

  // layer 2: WMMA GEMM -> propagate -> bias+relu   (bufB = h_final)
  gemm_wmma_f32<<<gGemm, T, 0, stream>>>(bufB, W2, bufA, N);
  zero_f32<<<gNH, T, 0, stream>>>(bufB, nh);
  edge_scatter<<<gEdge, T, 0, stream>>>(src, dst, dinv, bufA, bufB, E);
  finalize<<<gNH, T, 0, stream>>>(bufB, bufA, dinv, b2, bufB, N);

  // global mean pool
  int Bc = out_size / HID;
  zero_f32<<<(out_size + T - 1) / T, T, 0, stream>>>(out, out_size);
  zero_f32<<<1, T, 0, stream>>>(cnt, Bc);
  pool_cnt<<<gN, T, 0, stream>>>(batch, cnt, N);
  pool_accum<<<gNH, T, 0, stream>>>(bufB, batch, out, N);
  pool_scale<<<(out_size + T - 1) / T, T, 0, stream>>>(out, cnt, out_size);
}